// MaskedHead_29308856828594
// MI455X (gfx1250) — compile-verified
//
#include <hip/hip_runtime.h>

typedef __attribute__((ext_vector_type(2))) float v2f;
typedef __attribute__((ext_vector_type(8))) float v8f;

#define T_SEQ  2048
#define DMODEL 512
#define HDIM   64
#define NB     8
#define NROWS  (NB * T_SEQ)               // 16384
#define NTILES (T_SEQ / 16)               // 128
#define SCALE  0.04419417382415922f       // 512^-0.5

__device__ __forceinline__ v8f wmma4(v2f a, v2f b, v8f c) {
    // V_WMMA_F32_16X16X4_F32 : D = A(16x4) * B(4x16) + C(16x16), fp32
    return __builtin_amdgcn_wmma_f32_16x16x4_f32(false, a, false, b, (short)0, c,
                                                 false, false);
}

// ---------------- Kernel 1: Q/K/V projections -----------------------------
// Y[t,h] = sum_c X[t,c] * W[h,c]   (A = X rows, B[c,h] = W[h,c])
__global__ __launch_bounds__(128) void proj_kernel(const float* __restrict__ X,
                                                   const float* __restrict__ Wk,
                                                   const float* __restrict__ Wq,
                                                   const float* __restrict__ Wv,
                                                   float* __restrict__ ws) {
    const int wave  = threadIdx.x >> 5;
    const int lane  = threadIdx.x & 31;
    const int ln    = lane & 15;
    const int koff  = (lane < 16) ? 0 : 2;
    const int mbase = blockIdx.x * 16;
    const int nbase = wave * 16;                 // 4 waves cover HDIM=64
    const int which = blockIdx.y;                // 0=Q 1=K 2=V
    const float* W  = (which == 0) ? Wq : (which == 1) ? Wk : Wv;
    float* Y        = ws + (size_t)which * NROWS * HDIM;

    const float* xrow = X + (size_t)(mbase + ln) * DMODEL;
    const float* wrow = W + (size_t)(nbase + ln) * DMODEL;

    v8f acc = {};
    for (int c = 0; c < DMODEL; c += 4) {
        v2f a = *(const v2f*)(xrow + c + koff);  // A: K = c+koff, c+koff+1
        v2f b = *(const v2f*)(wrow + c + koff);  // B: rows K, cols nbase+ln
        acc = wmma4(a, b, acc);
    }
    const int rbase = (lane < 16) ? 0 : 8;
    float* yout = Y + (size_t)(mbase + rbase) * HDIM + nbase + ln;
#pragma unroll
    for (int r = 0; r < 8; ++r) yout[(size_t)r * HDIM] = acc[r];
}

// ---------------- Kernel 2: per-column softmax stats ----------------------
// softmax is over axis=1 (query axis i) => per column j: max/sum over i >= j
__global__ __launch_bounds__(32) void colstats_kernel(const float* __restrict__ ws,
                                                      float* __restrict__ Mcol,
                                                      float* __restrict__ Dcol) {
    const int lane  = threadIdx.x & 31;
    const int ln    = lane & 15;
    const int koff  = (lane < 16) ? 0 : 2;
    const int rbase = (lane < 16) ? 0 : 8;
    const int jt = blockIdx.x, b = blockIdx.y;
    const int jbase = jt * 16;

    const float* Q = ws + (size_t)b * T_SEQ * HDIM;
    const float* K = ws + (size_t)NROWS * HDIM + (size_t)b * T_SEQ * HDIM;
    const float* krow = K + (size_t)(jbase + ln) * HDIM;

    const float NEG_INF = -__builtin_inff();
    float m = NEG_INF, d = 0.f;

    for (int it = jt; it < NTILES; ++it) {       // causal: rows i >= j only
        const int ibase = it * 16;
        const float* qrow = Q + (size_t)(ibase + ln) * HDIM;
        v8f acc = {};
        for (int h = 0; h < HDIM; h += 4) {
            v2f a  = *(const v2f*)(qrow + h + koff);
            v2f bb = *(const v2f*)(krow + h + koff);
            acc = wmma4(a, bb, acc);
        }
        const int j = jbase + ln;
        float s[8];
        float mloc = NEG_INF;
#pragma unroll
        for (int r = 0; r < 8; ++r) {
            const int i = ibase + rbase + r;
            const float sv = (j <= i) ? acc[r] * SCALE : NEG_INF;
            s[r] = sv;
            mloc = fmaxf(mloc, sv);
        }
        const float mt   = fmaxf(mloc, __shfl_xor(mloc, 16, 32)); // join lane halves
        const float mnew = fmaxf(m, mt);
        float p = 0.f;
#pragma unroll
        for (int r = 0; r < 8; ++r) p += expf(s[r] - mnew);
        const float dt = p + __shfl_xor(p, 16, 32);
        d = d * expf(m - mnew) + dt;             // online softmax update
        m = mnew;
    }
    if (lane < 16) {
        Mcol[(size_t)b * T_SEQ + jbase + ln] = m;
        Dcol[(size_t)b * T_SEQ + jbase + ln] = d;
    }
}

// ---------------- Kernel 3: attn weights + output GEMM --------------------
__global__ __launch_bounds__(128) void attnout_kernel(const float* __restrict__ ws,
                                                      const float* __restrict__ Mcol,
                                                      const float* __restrict__ Dcol,
                                                      float* __restrict__ out) {
    __shared__ float sAttn[4][16][16];           // per-wave C->A layout staging
    __shared__ float sOut[4][16][64];            // cross-wave partial outputs

    const int wave  = threadIdx.x >> 5;
    const int lane  = threadIdx.x & 31;
    const int ln    = lane & 15;
    const int koff  = (lane < 16) ? 0 : 2;
    const int rbase = (lane < 16) ? 0 : 8;
    const int it = blockIdx.x, b = blockIdx.y;
    const int ibase = it * 16;

    const float* Q  = ws + (size_t)b * T_SEQ * HDIM;
    const float* K  = ws + (size_t)NROWS * HDIM + (size_t)b * T_SEQ * HDIM;
    const float* V  = ws + 2 * (size_t)NROWS * HDIM + (size_t)b * T_SEQ * HDIM;
    const float* Mc = Mcol + (size_t)b * T_SEQ;
    const float* Dc = Dcol + (size_t)b * T_SEQ;
    float* attnOut  = out + (size_t)NROWS * HDIM + (size_t)b * T_SEQ * T_SEQ;

    const float* qrow = Q + (size_t)(ibase + ln) * HDIM;

    v8f oacc[4] = {};                            // 16 rows x 64 cols of output

    for (int jt = wave; jt < NTILES; jt += 4) {  // waves split the column tiles
        const int jbase = jt * 16;
        const int j = jbase + ln;
        if (jt <= it) {
            const float* krow = K + (size_t)(jbase + ln) * HDIM;
            v8f acc = {};
            for (int h = 0; h < HDIM; h += 4) {
                v2f a  = *(const v2f*)(qrow + h + koff);
                v2f bb = *(const v2f*)(krow + h + koff);
                acc = wmma4(a, bb, acc);
            }
            const float mj  = Mc[j];
            const float inv = 1.0f / Dc[j];
            float wv[8];
#pragma unroll
            for (int r = 0; r < 8; ++r) {
                const int i = ibase + rbase + r;
                const float w = (j <= i) ? expf(acc[r] * SCALE - mj) * inv : 0.f;
                wv[r] = w;
                sAttn[wave][rbase + r][ln] = w;  // stage in C layout
            }
            float* arow = attnOut + (size_t)(ibase + rbase) * T_SEQ + j;
#pragma unroll
            for (int r = 0; r < 8; ++r) arow[(size_t)r * T_SEQ] = wv[r];

            // out[i,h] += attn(16x16) @ V[jbase:jbase+16, 0:64]
#pragma unroll
            for (int nt = 0; nt < 4; ++nt) {
                const int nbase = nt * 16;
#pragma unroll
                for (int k = 0; k < 16; k += 4) {
                    v2f a2, b2;
                    a2.x = sAttn[wave][ln][k + koff];       // A: M=ln, K
                    a2.y = sAttn[wave][ln][k + koff + 1];
                    b2.x = V[(size_t)(jbase + k + koff) * HDIM + nbase + ln];
                    b2.y = V[(size_t)(jbase + k + koff + 1) * HDIM + nbase + ln];
                    oacc[nt] = wmma4(a2, b2, oacc[nt]);
                }
            }
        } else {
            // strictly-upper tile: reference attn is exactly 0; d_out is poisoned
            float* arow = attnOut + (size_t)(ibase + rbase) * T_SEQ + j;
#pragma unroll
            for (int r = 0; r < 8; ++r) arow[(size_t)r * T_SEQ] = 0.f;
        }
    }

    // cross-wave reduction of the 16x64 output tile
#pragma unroll
    for (int nt = 0; nt < 4; ++nt)
#pragma unroll
        for (int r = 0; r < 8; ++r)
            sOut[wave][rbase + r][nt * 16 + ln] = oacc[nt][r];
    __syncthreads();
    for (int e = threadIdx.x; e < 16 * 64; e += 128) {
        const int row = e >> 6, h = e & 63;
        const float s = sOut[0][row][h] + sOut[1][row][h] +
                        sOut[2][row][h] + sOut[3][row][h];
        out[(size_t)(b * T_SEQ + ibase + row) * HDIM + h] = s;
    }
}

extern "C" void kernel_launch(void* const* d_in, const int* in_sizes, int n_in,
                              void* d_out, int out_size, void* d_ws, size_t ws_size,
                              hipStream_t stream) {
    const float* X  = (const float*)d_in[0];
    const float* Wk = (const float*)d_in[1];
    const float* Wq = (const float*)d_in[2];
    const float* Wv = (const float*)d_in[3];
    float* ws   = (float*)d_ws;
    float* outp = (float*)d_out;
    float* Mcol = ws + (size_t)3 * NROWS * HDIM;
    float* Dcol = Mcol + (size_t)NB * T_SEQ;

    proj_kernel<<<dim3(NROWS / 16, 3), 128, 0, stream>>>(X, Wk, Wq, Wv, ws);
    colstats_kernel<<<dim3(NTILES, NB), 32, 0, stream>>>(ws, Mcol, Dcol);
    attnout_kernel<<<dim3(NTILES, NB), 128, 0, stream>>>(ws, Mcol, Dcol, outp);
}